// TaskAlignedAssigner_71287867179416
// MI455X (gfx1250) — compile-verified
//
#include <hip/hip_runtime.h>

// Problem constants (from reference)
#define B_   16
#define N_   8400
#define NC_  80
#define M_   128
#define TOPK_ 10

constexpr int TILE = 128;                 // anchors per block in kernel 1
constexpr int SPAD = 85;                  // TDM-padded LDS row stride (floats): 80 data + 5 pads (1 DW per 16)
constexpr int NBLK = (N_ + TILE - 1) / TILE;  // 66 n-tiles per batch

typedef unsigned int u32;
typedef u32 v4u __attribute__((ext_vector_type(4)));
typedef u32 v8u __attribute__((ext_vector_type(8)));

// ---- CDNA5 async global->LDS copies (ASYNCcnt path) -----------------------
// LDS destination address = low 32 bits of the generic pointer (flat->LDS
// aperture truncates to [31:0]); verified in round-1 disasm.
__device__ __forceinline__ void async_g2l_b128(void* lds_dst, const void* gsrc) {
  unsigned lofs = (unsigned)(unsigned long long)lds_dst;
  asm volatile("global_load_async_to_lds_b128 %0, %1, off"
               :: "v"(lofs), "v"(gsrc) : "memory");
}
__device__ __forceinline__ void async_g2l_b32(void* lds_dst, const void* gsrc) {
  unsigned lofs = (unsigned)(unsigned long long)lds_dst;
  asm volatile("global_load_async_to_lds_b32 %0, %1, off"
               :: "v"(lofs), "v"(gsrc) : "memory");
}
__device__ __forceinline__ void wait_async0() {
  asm volatile("s_wait_asynccnt 0" ::: "memory");
}

// ---------------------------------------------------------------------------
// Kernel 1: am[b,m,n] = sqrt(score) * iou^6 ; also colmax[b,n] = max_m am.
// Score tile staged by ONE Tensor Data Mover descriptor (2-D tile, hardware
// padding to an 85-DWORD LDS row stride -> conflict-free per-m gathers).
// ---------------------------------------------------------------------------
__global__ __launch_bounds__(TILE) void am_colmax_kernel(
    const float* __restrict__ pred_scores,   // (B,N,NC)
    const float* __restrict__ pred_bboxes,   // (B,N,4)
    const int*   __restrict__ gt_labels,     // (B,M,1)
    const float* __restrict__ gt_bboxes,     // (B,M,4)
    float* __restrict__ out_am,              // (B,M,N)
    float* __restrict__ colmax)              // (B,N)
{
  __shared__ __align__(16) float  s_scores[TILE * SPAD];  // 43520 B
  __shared__ float4 s_gt[M_];                             // 2048 B
  __shared__ int    s_lab[M_];                            // 512 B

  const int tid = threadIdx.x;
  const int b   = blockIdx.x / NBLK;
  const int n0  = (blockIdx.x % NBLK) * TILE;
  const int n   = n0 + tid;
  const bool valid = (n < N_);
  const int nv  = (N_ - n0 < TILE) ? (N_ - n0) : TILE;

  // ---- stage GT boxes + labels via async DMA (tid spans exactly M_) ----
  async_g2l_b128(&s_gt[tid], (const float4*)gt_bboxes + (size_t)b * M_ + tid);
  async_g2l_b32(&s_lab[tid], gt_labels + (size_t)b * M_ + tid);

  // ---- stage score tile pred_scores[b, n0:n0+nv, 0:80] via TDM ----
  if (tid < 32) {                       // one wave issues the tensor op (EXEC ignored by TDM)
    const float* srcbase = pred_scores + ((size_t)b * N_ + n0) * NC_;
    unsigned long long ga = (unsigned long long)srcbase;
    u32 lds_base = (u32)(unsigned long long)(void*)&s_scores[0];
    lds_base = __builtin_amdgcn_readfirstlane(lds_base);

    // D# group 0: count=1 | lds_addr | global_addr[56:0] | type=2
    v4u g0;
    g0.x = 1u;
    g0.y = lds_base;
    g0.z = (u32)ga;
    g0.w = ((u32)(ga >> 32) & 0x01FFFFFFu) | 0x80000000u;

    // D# group 1: data_size=4B, pad every 16 DWORDs by 1 DWORD,
    // tensor/tile dim0 = 80, tensor/tile dim1 = nv, row stride = 80.
    v8u g1;
    g1.s0 = (2u << 16) | (1u << 20) | (3u << 22);  // data_size|pad_enable|pad_interval
    g1.s1 = (u32)NC_ << 16;                        // tensor_dim0[15:0] @ [63:48]
    g1.s2 = (u32)nv << 16;                         // tensor_dim1[15:0] @ [95:80]
    g1.s3 = (u32)NC_ << 16;                        // tile_dim0 @ [127:112]
    g1.s4 = (u32)nv;                               // tile_dim1 @ [143:128]
    g1.s5 = (u32)NC_;                              // tensor_dim0_stride[31:0]
    g1.s6 = 0u;
    g1.s7 = 0u;

    asm volatile("tensor_load_to_lds %0, %1" :: "s"(g0), "s"(g1) : "memory");
    __builtin_amdgcn_s_wait_tensorcnt(0);
  }

  // own pred box (regular vector load, coalesced b128)
  const int nclamp = valid ? n : (N_ - 1);
  float4 pb = ((const float4*)pred_bboxes)[(size_t)b * N_ + nclamp];
  const float parea = (pb.z - pb.x) * (pb.w - pb.y);

  wait_async0();
  __syncthreads();

  float vmax = -1.0f;
  const size_t obase = (size_t)b * M_ * N_ + n;   // index of (b, m=0, n)
  #pragma unroll 4
  for (int m = 0; m < M_; ++m) {
    float4 gb = s_gt[m];                // LDS broadcast
    int lab   = s_lab[m];               // LDS broadcast
    float il = fmaxf(pb.x, gb.x);
    float it = fmaxf(pb.y, gb.y);
    float ir = fminf(pb.z, gb.z);
    float ib = fminf(pb.w, gb.w);
    float w  = fmaxf(ir - il, 0.0f);
    float h  = fmaxf(ib - it, 0.0f);
    float inter = w * h;
    float garea = (gb.z - gb.x) * (gb.w - gb.y);
    float iou   = inter / (parea + garea - inter + 1e-16f);
    // TDM pad layout: column c lives at c + (c>>4) within an 85-DWORD row
    float sc    = s_scores[tid * SPAD + lab + (lab >> 4)];
    float i2    = iou * iou;
    float am    = sqrtf(sc) * (i2 * i2 * i2);     // score^0.5 * iou^6
    vmax = fmaxf(vmax, am);
    if (valid) out_am[obase + (size_t)m * N_] = am;
  }
  if (valid) colmax[(size_t)b * N_ + n] = vmax;
}

// ---------------------------------------------------------------------------
// Kernel 2: per (b,m) row of length N, exact top-10 indices -> scatter mask
// where am > 1e-9 and am == colmax[b,n]. One wave32 per row.
// ---------------------------------------------------------------------------
__global__ __launch_bounds__(32) void topk_scatter_kernel(
    const float* __restrict__ am,       // (B,M,N)
    const float* __restrict__ colmax,   // (B,N)
    float* __restrict__ mask)           // (B,M,N), pre-zeroed
{
  __shared__ float cv[32 * TOPK_];
  __shared__ int   cn[32 * TOPK_];

  const int b    = blockIdx.x / M_;
  const int m    = blockIdx.x % M_;
  const int lane = threadIdx.x;
  const float* row = am + ((size_t)b * M_ + m) * N_;

  // per-lane sorted (descending) top-10 over a strided slice of the row
  float v[TOPK_]; int ix[TOPK_];
  #pragma unroll
  for (int j = 0; j < TOPK_; ++j) { v[j] = -1.0f; ix[j] = 0x7fffffff; }
  for (int n = lane; n < N_; n += 32) {
    float x = row[n];
    if (x > v[TOPK_ - 1]) {             // strict > keeps first (lowest-n) on ties
      v[TOPK_ - 1] = x; ix[TOPK_ - 1] = n;
      #pragma unroll
      for (int j = TOPK_ - 1; j > 0; --j) {
        bool sw = (v[j] > v[j - 1]);
        if (sw) {
          float tv = v[j]; v[j] = v[j - 1]; v[j - 1] = tv;
          int   ti = ix[j]; ix[j] = ix[j - 1]; ix[j - 1] = ti;
        }
      }
    }
  }
  #pragma unroll
  for (int j = 0; j < TOPK_; ++j) { cv[lane * TOPK_ + j] = v[j]; cn[lane * TOPK_ + j] = ix[j]; }
  __syncthreads();

  // 10 rounds of wave-wide argmax over the 320 candidates, invalidate winner
  const size_t mbase = ((size_t)b * M_ + m) * N_;
  for (int k = 0; k < TOPK_; ++k) {
    float bv = -2.0f; int bn = 0x7fffffff; int bs = 0;
    #pragma unroll
    for (int j = 0; j < TOPK_; ++j) {
      int s = lane * TOPK_ + j;
      float x = cv[s]; int nn = cn[s];
      if (x > bv || (x == bv && nn < bn)) { bv = x; bn = nn; bs = s; }
    }
    #pragma unroll
    for (int off = 16; off > 0; off >>= 1) {   // wave32 butterfly (lowest-n tiebreak)
      float ov = __shfl_xor(bv, off, 32);
      int   on = __shfl_xor(bn, off, 32);
      int   os = __shfl_xor(bs, off, 32);
      if (ov > bv || (ov == bv && on < bn)) { bv = ov; bn = on; bs = os; }
    }
    if (lane == 0) {
      cv[bs] = -2.0f;                          // remove winner
      if (bv > 1e-9f && bv == colmax[(size_t)b * N_ + bn])
        mask[mbase + bn] = 1.0f;
    }
    __syncthreads();
  }
}

// ---------------------------------------------------------------------------
extern "C" void kernel_launch(void* const* d_in, const int* in_sizes, int n_in,
                              void* d_out, int out_size, void* d_ws, size_t ws_size,
                              hipStream_t stream) {
  const float* pred_scores = (const float*)d_in[0];
  const float* pred_bboxes = (const float*)d_in[1];
  const int*   gt_labels   = (const int*)  d_in[2];
  const float* gt_bboxes   = (const float*)d_in[3];

  const size_t plane = (size_t)B_ * M_ * N_;
  float* out_mask = (float*)d_out;           // output 0: mask_pos (B,M,N)
  float* out_am   = (float*)d_out + plane;   // output 1: am       (B,M,N)
  float* ws_colmax = (float*)d_ws;           // (B,N) column max over m

  hipMemsetAsync(out_mask, 0, plane * sizeof(float), stream);

  am_colmax_kernel<<<dim3(B_ * NBLK), dim3(TILE), 0, stream>>>(
      pred_scores, pred_bboxes, gt_labels, gt_bboxes, out_am, ws_colmax);

  topk_scatter_kernel<<<dim3(B_ * M_), dim3(32), 0, stream>>>(
      out_am, ws_colmax, out_mask);
}